// SuperLoss_36833639530989
// MI455X (gfx1250) — compile-verified
//
#include <hip/hip_runtime.h>
#include <math.h>
#include <cmath>

typedef float v2f __attribute__((ext_vector_type(2)));
typedef float v4f __attribute__((ext_vector_type(4)));
typedef float v8f __attribute__((ext_vector_type(8)));

#define NEG_INF (-__builtin_inff())

// Combine two online-logsumexp states (m = running max, s = sum of exp(x - m)).
// Guarded so that combining two empty states (-inf, 0) stays (-inf, 0) instead
// of producing NaN via exp(-inf - -inf).
__device__ __forceinline__ void online_combine(float& m, float& s, float mo, float so) {
    float mn = fmaxf(m, mo);
    float ea = (mn == NEG_INF) ? 0.0f : __expf(m - mn);
    float eb = (mn == NEG_INF) ? 0.0f : __expf(mo - mn);
    s = s * ea + so * eb;
    m = mn;
}

// Online update with one float4 chunk: one rescale exp + 4 element exps.
__device__ __forceinline__ void online_chunk(float& m, float& s, v4f v) {
    float cmax = fmaxf(fmaxf(v.x, v.y), fmaxf(v.z, v.w));
    float mn = fmaxf(m, cmax);
    float sc = s * __expf(m - mn);   // exp(-inf)=0 on first chunk: safe
    sc += __expf(v.x - mn);
    sc += __expf(v.y - mn);
    sc += __expf(v.z - mn);
    sc += __expf(v.w - mn);
    s = sc;
    m = mn;
}

// Kernel 1 — one block per row: streaming online logsumexp, write per-row CE.
// 4 independent online states per thread over 4 strided b128 NT loads per
// iteration: batches loads for MLP and breaks the serial exp/accumulate chain.
__global__ __launch_bounds__(256) void ce_row_kernel(
    const float* __restrict__ logits, const int* __restrict__ targets,
    float* __restrict__ ce_out, int C)
{
    const int row = blockIdx.x;
    const int tid = threadIdx.x;
    const float* rowp = logits + (size_t)row * (size_t)C;

    // Rows are not 16B-aligned (C % 4 == 1): peel to alignment, stream b128,
    // small scalar tail.
    int prefix = (int)(((16u - ((unsigned)(uintptr_t)rowp & 15u)) & 15u) >> 2);
    if (prefix > C) prefix = C;
    const int n4   = (C - prefix) >> 2;
    const int tail = (C - prefix) & 3;

    float m0 = NEG_INF, m1 = NEG_INF, m2 = NEG_INF, m3 = NEG_INF;
    float s0 = 0.0f,    s1 = 0.0f,    s2 = 0.0f,    s3 = 0.0f;

    const int nscalar = prefix + tail;  // <= 6, one element per low thread
    if (tid < nscalar) {
        int idx = (tid < prefix) ? tid : (prefix + 4 * n4 + (tid - prefix));
        m0 = rowp[idx];
        s0 = 1.0f;
    }

    const v4f* p4 = (const v4f*)(rowp + prefix);
    const int step = 256;
    int i = tid;
    for (; i + 3 * step < n4; i += 4 * step) {
        v4f a = __builtin_nontemporal_load(p4 + i);
        v4f b = __builtin_nontemporal_load(p4 + i + step);
        v4f c = __builtin_nontemporal_load(p4 + i + 2 * step);
        v4f d = __builtin_nontemporal_load(p4 + i + 3 * step);
        online_chunk(m0, s0, a);
        online_chunk(m1, s1, b);
        online_chunk(m2, s2, c);
        online_chunk(m3, s3, d);
    }
    for (; i < n4; i += step) {
        v4f a = __builtin_nontemporal_load(p4 + i);
        online_chunk(m0, s0, a);
    }
    online_combine(m0, s0, m1, s1);
    online_combine(m2, s2, m3, s3);
    online_combine(m0, s0, m2, s2);

    // Wave32 reduction.
    #pragma unroll
    for (int off = 16; off >= 1; off >>= 1) {
        float mo = __shfl_xor(m0, off);
        float so = __shfl_xor(s0, off);
        online_combine(m0, s0, mo, so);
    }

    __shared__ float sh_m[8];
    __shared__ float sh_s[8];
    const int wid = tid >> 5;
    if ((tid & 31) == 0) { sh_m[wid] = m0; sh_s[wid] = s0; }
    __syncthreads();

    if (tid == 0) {
        float M = sh_m[0], S = sh_s[0];
        #pragma unroll
        for (int w = 1; w < 8; ++w) online_combine(M, S, sh_m[w], sh_s[w]);
        float lse = M + logf(S);
        ce_out[row] = lse - rowp[targets[row]];
    }
}

// Kernel 2 — per-row SuperLoss from CE (all Lambert-W Halley loops in parallel).
__global__ __launch_bounds__(256) void superloss_kernel(
    const float* __restrict__ ce_in, float* __restrict__ loss_out,
    int B, float tau, float lam)
{
    const int i = blockIdx.x * blockDim.x + threadIdx.x;
    if (i >= B) return;

    const float ce = ce_in[i];
    const float INV_E = 0.36787944117144233f;
    const float E     = 2.7182818284590452f;

    float y = 0.5f * fmaxf(-2.0f * INV_E, (ce - tau) / lam);
    y = fmaxf(y, -INV_E + 1e-7f);

    float p  = sqrtf(fmaxf(2.0f * (E * y + 1.0f), 0.0f));
    float wn = -1.0f + p - p * p * (1.0f / 3.0f);
    float w  = (y < 0.5f) ? wn : log1pf(y);
    #pragma unroll 1
    for (int it = 0; it < 20; ++it) {
        float ew = expf(w);
        float f  = w * ew - y;
        float denom = ew * (w + 1.0f) - (w + 2.0f) * f / (2.0f * w + 2.0f);
        w = w - f / denom;
    }

    float sigma = expf(-w);
    float lg    = logf(sigma);
    loss_out[i] = (ce - tau) * sigma + lam * lg * lg;
}

// Kernel 3 — one full wave (EXEC all-ones, as WMMA requires).
// V_WMMA_F32_16X16X4_F32 with B = ones reduces the 32 lane partials:
//   A[m][0]=p_m (lanes 0-15, K=0), A[m][2]=p_{m+16} (lanes 16-31, K=2)
//   => D[m][n] = p_m + p_{m+16} for every n.
// A lane's 8 D VGPRs sum rows 0-7 (lanes<16) or 8-15 (lanes>=16); adding
// lane 0 + lane 16 yields the exact total of all 32 partials.
__global__ __launch_bounds__(32) void reduce_mean_wmma_kernel(
    const float* __restrict__ vals, float* __restrict__ out, int B)
{
    const int lane = threadIdx.x;
    float p = 0.0f;
    for (int i = lane; i < B; i += 32) p += vals[i];

    v2f a; a[0] = p;    a[1] = 0.0f;
    v2f b; b[0] = 1.0f; b[1] = 1.0f;
    v8f c = {};
    v8f d = __builtin_amdgcn_wmma_f32_16x16x4_f32(
        /*neg_a=*/false, a, /*neg_b=*/false, b,
        /*c_mod=*/(short)0, c, /*reuse_a=*/false, /*reuse_b=*/false);

    float t = d[0] + d[1] + d[2] + d[3] + d[4] + d[5] + d[6] + d[7];
    float total = __shfl(t, 0) + __shfl(t, 16);
    if (lane == 0) out[0] = total / (float)B;
}

extern "C" void kernel_launch(void* const* d_in, const int* in_sizes, int n_in,
                              void* d_out, int out_size, void* d_ws, size_t ws_size,
                              hipStream_t stream) {
    const float* logits  = (const float*)d_in[0];
    const int*   targets = (const int*)d_in[1];
    const int B = in_sizes[1];
    const int C = in_sizes[0] / B;
    const float tau = (float)log((double)C);
    const float lam = 1.0f;

    float* per_row_ce   = (float*)d_ws;       // B floats
    float* per_row_loss = per_row_ce + B;     // B floats

    ce_row_kernel<<<B, 256, 0, stream>>>(logits, targets, per_row_ce, C);
    superloss_kernel<<<(B + 255) / 256, 256, 0, stream>>>(per_row_ce, per_row_loss,
                                                          B, tau, lam);
    reduce_mean_wmma_kernel<<<1, 32, 0, stream>>>(per_row_loss, (float*)d_out, B);
}